// YOLOLoss_52647709114896
// MI455X (gfx1250) — compile-verified
//
#include <hip/hip_runtime.h>

// ---- problem constants (match reference) ----
#define S_DIM   56
#define C_CLS   80
#define D_CH    90                       // B*5 + C
#define NB      64
#define TOT     (NB * S_DIM * S_DIM)     // 200704
#define BLK     256
#define GRID1   (TOT / BLK)              // 784 (exact)
#define L_COORD 5.0f
#define L_OBJ   1.0f
#define L_NOOBJ 0.5f
#define L_CLASS 1.0f

typedef __attribute__((ext_vector_type(2))) float v2f;
typedef __attribute__((ext_vector_type(8))) float v8f;

// Exact 32-lane sum on the matrix pipe:
// A.v0 = v, A.v1 = 0, B = ones  ->  D[m,n] = v[m] + v[m+16]  (n arbitrary)
// Sum D's 8 VGPRs per lane, then fold halves: every lane holds the full sum.
__device__ __forceinline__ float wave_sum32(float v) {
    v2f a; a.x = v;    a.y = 0.0f;
    v2f b; b.x = 1.0f; b.y = 1.0f;
    v8f c = {};
    c = __builtin_amdgcn_wmma_f32_16x16x4_f32(false, a, false, b,
                                              (short)0, c, false, false);
    float r = ((c[0] + c[1]) + (c[2] + c[3])) + ((c[4] + c[5]) + (c[6] + c[7]));
    r += __shfl_xor(r, 16, 32);
    return r;
}

__device__ __forceinline__ int wave_max32(int v) {
    #pragma unroll
    for (int off = 16; off > 0; off >>= 1) {
        int o = __shfl_xor(v, off, 32);
        v = (o > v) ? o : v;
    }
    return v;
}

// Row-wise IoU of center-format boxes, op-for-op like the reference.
__device__ __forceinline__ float iou_ref(float ax, float ay, float aw, float ah,
                                         float bx, float by, float bw, float bh) {
    float ax1 = ax - aw * 0.5f, ay1 = ay - ah * 0.5f;
    float ax2 = ax + aw * 0.5f, ay2 = ay + ah * 0.5f;
    float bx1 = bx - bw * 0.5f, by1 = by - bh * 0.5f;
    float bx2 = bx + bw * 0.5f, by2 = by + bh * 0.5f;
    float iw = fmaxf(fminf(ax2, bx2) - fmaxf(ax1, bx1), 0.0f);
    float ih = fmaxf(fminf(ay2, by2) - fmaxf(ay1, by1), 0.0f);
    float inter = iw * ih;
    float uni = (ax2 - ax1) * (ay2 - ay1) + (bx2 - bx1) * (by2 - by1) - inter;
    return inter / (uni + 1e-6f);
}

__global__ __launch_bounds__(BLK)
void yolo_stage1(const float* __restrict__ preds,
                 const float* __restrict__ targets,
                 float* __restrict__ partial,     // [GRID1][3]
                 int*   __restrict__ lastBlk) {   // [GRID1]
    const int i    = blockIdx.x * BLK + threadIdx.x;      // always < TOT
    const int cell = i % (S_DIM * S_DIM);
    const float col = (float)(cell % S_DIM);
    const float row = (float)(cell / S_DIM);

    const float* p = preds   + (size_t)i * D_CH;
    const float* t = targets + (size_t)i * D_CH;

    // only the 10 live channels per tensor (5 x b64 loads each, 8B aligned)
    float pv[10], tv[10];
    #pragma unroll
    for (int j = 0; j < 5; ++j) {
        v2f a = *reinterpret_cast<const v2f*>(p + 2 * j);
        v2f b = *reinterpret_cast<const v2f*>(t + 2 * j);
        pv[2 * j] = a.x; pv[2 * j + 1] = a.y;
        tv[2 * j] = b.x; tv[2 * j + 1] = b.y;
    }

    const float fS = (float)S_DIM;
    // decode (cx,cy image-relative; w,h raw; conf)
    float px0 = (pv[0] + col) / fS, py0 = (pv[1] + row) / fS;
    float pw0 = pv[2], ph0 = pv[3], pc0 = pv[4];
    float px1 = (pv[5] + col) / fS, py1 = (pv[6] + row) / fS;
    float pw1 = pv[7], ph1 = pv[8], pc1 = pv[9];
    float tx0 = (tv[0] + col) / fS, ty0 = (tv[1] + row) / fS;
    float tw0 = tv[2], th0 = tv[3], tc0 = tv[4];
    float tx1 = (tv[5] + col) / fS, ty1 = (tv[6] + row) / fS;
    float tw1 = tv[7], th1 = tv[8], tc1 = tv[9];

    float iou0 = iou_ref(px0, py0, pw0, ph0, tx0, ty0, tw0, th0);
    float iou1 = iou_ref(px1, py1, pw1, ph1, tx1, ty1, tw1, th1);

    const bool  b1 = iou1 > iou0;           // argmax (first-max tie rule)
    const float m  = (tc0 > 0.0f) ? 1.0f : 0.0f;

    // responsible-box coord + obj terms (cndmask selects; no divergence)
    float dx = (b1 ? px1 : px0) - (b1 ? tx1 : tx0);
    float dy = (b1 ? py1 : py0) - (b1 ? ty1 : ty0);
    float dw = (b1 ? pw1 : pw0) - (b1 ? tw1 : tw0);
    float dh = (b1 ? ph1 : ph0) - (b1 ? th1 : th0);
    float coord = m * (dx * dx + dy * dy + dw * dw + dh * dh);

    float dc  = (b1 ? pc1 : pc0) - (b1 ? tc1 : tc0);
    float obj = m * dc * dc;

    float e0 = pc0 - tc0, e1 = pc1 - tc1;
    float pcn = b1 ? pc0 : pc1;             // non-responsible conf
    float noobj = (1.0f - m) * (e0 * e0 + e1 * e1) + m * (pcn * pcn);

    int cand = (tc0 > 0.0f) ? i : -1;       // for last-obj-cell (class loss)

    // wave reduction on the matrix pipe, then cross-wave via LDS
    float wc = wave_sum32(coord);
    float wo = wave_sum32(obj);
    float wn = wave_sum32(noobj);
    int   wi = wave_max32(cand);

    __shared__ float sC[8], sO[8], sN[8];
    __shared__ int   sI[8];
    const int wave = threadIdx.x >> 5, lane = threadIdx.x & 31;
    if (lane == 0) { sC[wave] = wc; sO[wave] = wo; sN[wave] = wn; sI[wave] = wi; }
    __syncthreads();
    if (threadIdx.x == 0) {
        float c = 0.f, o = 0.f, n = 0.f; int mi = -1;
        #pragma unroll
        for (int w = 0; w < 8; ++w) {
            c += sC[w]; o += sO[w]; n += sN[w];
            mi = (sI[w] > mi) ? sI[w] : mi;
        }
        partial[blockIdx.x * 3 + 0] = c;
        partial[blockIdx.x * 3 + 1] = o;
        partial[blockIdx.x * 3 + 2] = n;
        lastBlk[blockIdx.x] = mi;
    }
}

__global__ __launch_bounds__(BLK)
void yolo_stage2(const float* __restrict__ preds,
                 const float* __restrict__ targets,
                 const float* __restrict__ partial,
                 const int*   __restrict__ lastBlk,
                 float* __restrict__ out) {
    float c = 0.f, o = 0.f, n = 0.f; int mi = -1;
    for (int j = threadIdx.x; j < GRID1; j += BLK) {
        c += partial[j * 3 + 0];
        o += partial[j * 3 + 1];
        n += partial[j * 3 + 2];
        int v = lastBlk[j];
        mi = (v > mi) ? v : mi;
    }
    float wc = wave_sum32(c), wo = wave_sum32(o), wn = wave_sum32(n);
    int   wi = wave_max32(mi);

    __shared__ float sC[8], sO[8], sN[8];
    __shared__ int   sI[8];
    __shared__ float sSum[3];
    __shared__ int   sLast;
    const int wave = threadIdx.x >> 5, lane = threadIdx.x & 31;
    if (lane == 0) { sC[wave] = wc; sO[wave] = wo; sN[wave] = wn; sI[wave] = wi; }
    __syncthreads();
    if (threadIdx.x == 0) {
        float cc = 0.f, oo = 0.f, nn = 0.f; int last = -1;
        #pragma unroll
        for (int w = 0; w < 8; ++w) {
            cc += sC[w]; oo += sO[w]; nn += sN[w];
            last = (sI[w] > last) ? sI[w] : last;
        }
        if (last < 0) last = TOT - 1;       // JAX index -1 wraps to last row
        sSum[0] = cc; sSum[1] = oo; sSum[2] = nn; sLast = last;
    }
    __syncthreads();

    // class loss of the last object cell: channels [10, 90)
    const int last = sLast;
    const int ch   = (threadIdx.x < C_CLS) ? threadIdx.x : (C_CLS - 1); // clamp load
    const size_t base = (size_t)last * D_CH + 10;
    float d   = preds[base + ch] - targets[base + ch];
    float val = (threadIdx.x < C_CLS) ? d * d : 0.0f;   // mask, EXEC stays full
    float wcl = wave_sum32(val);
    if (lane == 0) sC[wave] = wcl;
    __syncthreads();
    if (threadIdx.x == 0) {
        float cls = 0.f;
        #pragma unroll
        for (int w = 0; w < 8; ++w) cls += sC[w];
        float coord_l = L_COORD * sSum[0];
        float obj_l   = L_OBJ   * sSum[1];
        float noobj_l = L_NOOBJ * sSum[2];
        float cls_l   = L_CLASS * cls;
        float total   = coord_l + obj_l + noobj_l + cls_l;
        out[0] = total / (float)NB;
        out[1] = coord_l;
        out[2] = obj_l;
        out[3] = noobj_l;
        out[4] = cls_l;
    }
}

extern "C" void kernel_launch(void* const* d_in, const int* in_sizes, int n_in,
                              void* d_out, int out_size, void* d_ws, size_t ws_size,
                              hipStream_t stream) {
    const float* preds   = (const float*)d_in[0];
    const float* targets = (const float*)d_in[1];
    float* partial = (float*)d_ws;                                   // GRID1*3 floats
    int*   lastBlk = (int*)((char*)d_ws + GRID1 * 3 * sizeof(float)); // GRID1 ints
    float* out = (float*)d_out;

    yolo_stage1<<<GRID1, BLK, 0, stream>>>(preds, targets, partial, lastBlk);
    yolo_stage2<<<1, BLK, 0, stream>>>(preds, targets, partial, lastBlk, out);
}